// GraphormerEncoderLayer_45844480918019
// MI455X (gfx1250) — compile-verified
//
#include <hip/hip_runtime.h>
#include <hip/hip_bf16.h>
#include <math.h>

typedef __attribute__((ext_vector_type(2))) float v2f;
typedef __attribute__((ext_vector_type(8))) float v8f;

#define Bb 16
#define Nn 512
#define Dd 768
#define Hh 12
#define Ff 3072
#define HD 64

__device__ __forceinline__ float gelu_tanh(float x) {
    float x3 = x * x * x;
    return 0.5f * x * (1.0f + tanhf(0.7978845608028654f * (x + 0.044715f * x3)));
}

// ---------------------------------------------------------------------------
// Kernel 1: fused double LayerNorm.  One block per (b,n) row; one mean/var,
// two affine outputs: h1 = LN(x)*g1+be1 and h2 = LN(x)*g2+be2.
// ---------------------------------------------------------------------------
__global__ __launch_bounds__(256)
void ln2_kernel(const float* __restrict__ x,
                const float* __restrict__ g1, const float* __restrict__ be1,
                const float* __restrict__ g2, const float* __restrict__ be2,
                float* __restrict__ h1, float* __restrict__ h2) {
    __shared__ float red[256];
    const int row = blockIdx.x;                 // 0 .. B*N-1
    const int tid = threadIdx.x;
    const float* xr = x + (size_t)row * Dd;

    float s = 0.0f;
    for (int i = tid; i < Dd; i += 256) s += xr[i];
    red[tid] = s; __syncthreads();
    for (int off = 128; off > 0; off >>= 1) {
        if (tid < off) red[tid] += red[tid + off];
        __syncthreads();
    }
    const float mu = red[0] * (1.0f / Dd);
    __syncthreads();

    float v = 0.0f;
    for (int i = tid; i < Dd; i += 256) { float d = xr[i] - mu; v += d * d; }
    red[tid] = v; __syncthreads();
    for (int off = 128; off > 0; off >>= 1) {
        if (tid < off) red[tid] += red[tid + off];
        __syncthreads();
    }
    const float rstd = rsqrtf(red[0] * (1.0f / Dd) + 1e-5f);

    for (int i = tid; i < Dd; i += 256) {
        float nh = (xr[i] - mu) * rstd;
        h1[(size_t)row * Dd + i] = nh * g1[i] + be1[i];
        h2[(size_t)row * Dd + i] = nh * g2[i] + be2[i];
    }
}

// ---------------------------------------------------------------------------
// Kernel 2: register-blocked f32 WMMA GEMM, C[M,N] = A[M,K]*B[K,N] + epilogue.
// One wave owns a 64x32 C macro-tile: 4 M-tiles x 2 N-tiles = 8 accumulators.
// Per K-step (V_WMMA_F32_16X16X4_F32): 4 contiguous b64 A loads + 4 b32 B
// loads feed 8 WMMAs -> ~1 vmem issue per WMMA (vs 3 in the naive version).
// EPI: 0 = (+bias), 1 = +bias+resid, 2 = gelu(+bias), 3 = +bias+resid+mask0.
// ---------------------------------------------------------------------------
template <int EPI>
__global__ __launch_bounds__(256)
void gemm_kernel(const float* __restrict__ A, const float* __restrict__ Bm,
                 const float* __restrict__ bias, const float* __restrict__ resid,
                 const float* __restrict__ mask, float* __restrict__ C,
                 int M, int N, int K) {
    const int nblk = N >> 5;                              // 32-col macro tiles
    const int gw = blockIdx.x * 8 + (threadIdx.x >> 5);   // global wave id
    const int mb = gw / nblk;
    const int nb = gw - mb * nblk;
    if (mb >= (M >> 6)) return;                           // wave-uniform exit

    const int l  = threadIdx.x & 31;
    const int lm = l & 15;            // row (A) / col (B,C) within 16x16 tile
    const int hi = l >> 4;            // lane half selects K pair {0,1} vs {2,3}
    const int m0 = mb << 6;
    const int n0 = nb << 5;

    const float* arow0 = A  + (size_t)(m0 + lm) * K + 2 * hi;   // 8B aligned
    const float* bcol0 = Bm + (size_t)(2 * hi) * N + n0 + lm;

    v8f acc[4][2];
#pragma unroll
    for (int i = 0; i < 4; ++i)
#pragma unroll
        for (int j = 0; j < 2; ++j) acc[i][j] = (v8f){};

    for (int k = 0; k < K; k += 4) {
        v2f a[4];
#pragma unroll
        for (int i = 0; i < 4; ++i)
            a[i] = *(const v2f*)(arow0 + (size_t)(16 * i) * K + k);
        v2f bf[2];
#pragma unroll
        for (int j = 0; j < 2; ++j) {
            bf[j].x = bcol0[(size_t)k * N + 16 * j];
            bf[j].y = bcol0[(size_t)(k + 1) * N + 16 * j];
        }
#pragma unroll
        for (int i = 0; i < 4; ++i)
#pragma unroll
            for (int j = 0; j < 2; ++j)
                acc[i][j] = __builtin_amdgcn_wmma_f32_16x16x4_f32(
                    false, a[i], false, bf[j], (short)0, acc[i][j], false, false);
    }

#pragma unroll
    for (int j = 0; j < 2; ++j) {
        const int col = n0 + 16 * j + lm;
        const float bvv = bias ? bias[col] : 0.0f;
#pragma unroll
        for (int i = 0; i < 4; ++i) {
#pragma unroll
            for (int r = 0; r < 8; ++r) {
                const int row = m0 + 16 * i + r + 8 * hi;
                float v = acc[i][j][r] + bvv;
                if (EPI == 1 || EPI == 3) v += resid[(size_t)row * N + col];
                if (EPI == 2) v = gelu_tanh(v);
                if (EPI == 3) { if (mask[row] != 0.0f) v = 0.0f; }
                C[(size_t)row * N + col] = v;
            }
        }
    }
}

// ---------------------------------------------------------------------------
// Kernel 3: fused attention for one (b, h, 16-query tile).
//   S = Q K^T * scale + (spatial+edge)   -> 16x512 LDS tile (WMMA, Q fragment
//                                           reused across 4 key tiles)
//   row-nonzero -> -inf semantics; wave-cooperative softmax (__shfl_xor)
//   O = P V                              -> WMMA; one ds_load_b64 of P per
//                                           K-step shared by all 4 d-tiles
// Single wave per block; EXEC all-ones for every WMMA.
// ---------------------------------------------------------------------------
__global__ __launch_bounds__(32)
void attn_kernel(const float* __restrict__ q, const float* __restrict__ kbuf,
                 const float* __restrict__ vbuf, const float* __restrict__ sp,
                 const float* __restrict__ ed, float* __restrict__ obuf) {
    __shared__ float smem[16][Nn];               // 32 KB score/prob tile

    const int blk = blockIdx.x;                  // B*H*(N/16) blocks
    const int qt = blk & 31;                     // N/16 = 32 query tiles
    const int h  = (blk >> 5) % Hh;
    const int b  = blk / (32 * Hh);
    const int l  = threadIdx.x;
    const int lm = l & 15;
    const int hi = l >> 4;
    const int q0 = qt << 4;
    const float scale = 0.036084391824351615f;   // 768^-0.5

    const size_t rowQ = ((size_t)(b * Nn + q0 + lm)) * Dd + h * HD + 2 * hi;

    // ---- scores: S[16,512], 4 key tiles per pass ---------------------------
    for (int ng = 0; ng < 8; ++ng) {             // groups of 4 x 16 keys
        const int n0 = ng << 6;
        v8f acc[4];
#pragma unroll
        for (int j = 0; j < 4; ++j) acc[j] = (v8f){};
        size_t rowK[4];
#pragma unroll
        for (int j = 0; j < 4; ++j)
            rowK[j] = ((size_t)(b * Nn + n0 + 16 * j + lm)) * Dd + h * HD + 2 * hi;
#pragma unroll
        for (int kk = 0; kk < HD; kk += 4) {
            v2f a = *(const v2f*)(q + rowQ + kk);          // 1 load, 4 uses
#pragma unroll
            for (int j = 0; j < 4; ++j) {
                v2f bb = *(const v2f*)(kbuf + rowK[j] + kk);
                acc[j] = __builtin_amdgcn_wmma_f32_16x16x4_f32(
                    false, a, false, bb, (short)0, acc[j], false, false);
            }
        }
#pragma unroll
        for (int j = 0; j < 4; ++j) {
#pragma unroll
            for (int r = 0; r < 8; ++r) {
                const int row = r + 8 * hi;
                const int col = n0 + 16 * j + lm;
                const size_t bidx = ((size_t)b * Nn + (q0 + row)) * Nn + col;
                smem[row][col] = acc[j][r] * scale + sp[bidx] + ed[bidx];
            }
        }
    }

    // ---- softmax with all-zero-row -> zero-prob semantics ------------------
    for (int r = 0; r < 16; ++r) {
        float vals[16];
        float vmax = -3.0e38f;
        int nz = 0;
#pragma unroll
        for (int i = 0; i < 16; ++i) {
            float t = smem[r][l + 32 * i];
            vals[i] = t;
            nz |= (t != 0.0f);
            vmax = fmaxf(vmax, t);
        }
        for (int off = 16; off > 0; off >>= 1)
            vmax = fmaxf(vmax, __shfl_xor(vmax, off, 32));
        const unsigned long long anynz = __ballot(nz);
        float sum = 0.0f;
        if (anynz) {
#pragma unroll
            for (int i = 0; i < 16; ++i) {
                float e = __expf(vals[i] - vmax);
                vals[i] = e;
                sum += e;
            }
        }
        for (int off = 16; off > 0; off >>= 1)
            sum += __shfl_xor(sum, off, 32);
        const float inv = anynz ? (1.0f / sum) : 0.0f;
#pragma unroll
        for (int i = 0; i < 16; ++i)
            smem[r][l + 32 * i] = vals[i] * inv;
    }

    // ---- O = P V : one LDS A-fragment per K-step, 4 d-tiles ----------------
    {
        v8f acc[4];
#pragma unroll
        for (int dt = 0; dt < 4; ++dt) acc[dt] = (v8f){};
        for (int kk = 0; kk < Nn; kk += 4) {
            v2f a = *(const v2f*)(&smem[lm][kk + 2 * hi]);      // ds_load_b64
            const size_t vb =
                ((size_t)(b * Nn + kk + 2 * hi)) * Dd + h * HD + lm;
#pragma unroll
            for (int dt = 0; dt < 4; ++dt) {
                v2f bb;
                bb.x = vbuf[vb + 16 * dt];
                bb.y = vbuf[vb + 16 * dt + Dd];
                acc[dt] = __builtin_amdgcn_wmma_f32_16x16x4_f32(
                    false, a, false, bb, (short)0, acc[dt], false, false);
            }
        }
#pragma unroll
        for (int dt = 0; dt < 4; ++dt) {
#pragma unroll
            for (int r = 0; r < 8; ++r) {
                const int row = r + 8 * hi;
                obuf[((size_t)(b * Nn + q0 + row)) * Dd + h * HD + 16 * dt + lm] =
                    acc[dt][r];
            }
        }
    }
}

// ---------------------------------------------------------------------------
// Kernel 4: pad mask — any(att[b,n,:] == 0) per row.
// ---------------------------------------------------------------------------
__global__ __launch_bounds__(256)
void mask_kernel(const float* __restrict__ att, float* __restrict__ mask) {
    __shared__ int any0;
    const int row = blockIdx.x;
    if (threadIdx.x == 0) any0 = 0;
    __syncthreads();
    int loc = 0;
    for (int i = threadIdx.x; i < Dd; i += 256)
        loc |= (att[(size_t)row * Dd + i] == 0.0f);
    if (loc) atomicOr(&any0, 1);
    __syncthreads();
    if (threadIdx.x == 0) mask[row] = any0 ? 1.0f : 0.0f;
}

// ---------------------------------------------------------------------------
extern "C" void kernel_launch(void* const* d_in, const int* in_sizes, int n_in,
                              void* d_out, int out_size, void* d_ws, size_t ws_size,
                              hipStream_t stream) {
    const float* x   = (const float*)d_in[0];
    const float* sp  = (const float*)d_in[1];
    const float* ed  = (const float*)d_in[2];
    const float* Wq  = (const float*)d_in[3];
    const float* Wk  = (const float*)d_in[4];
    const float* Wv  = (const float*)d_in[5];
    const float* bv  = (const float*)d_in[6];
    const float* Wo  = (const float*)d_in[7];
    const float* bo  = (const float*)d_in[8];
    const float* g1  = (const float*)d_in[9];
    const float* be1 = (const float*)d_in[10];
    const float* g2  = (const float*)d_in[11];
    const float* be2 = (const float*)d_in[12];
    const float* W1  = (const float*)d_in[13];
    const float* bf1 = (const float*)d_in[14];
    const float* W2  = (const float*)d_in[15];
    const float* bf2 = (const float*)d_in[16];
    float* out = (float*)d_out;

    const size_t BND = (size_t)Bb * Nn * Dd;   // 6,291,456 floats
    const size_t BNF = (size_t)Bb * Nn * Ff;   // 25,165,824 floats
    float* ws    = (float*)d_ws;
    float* h1    = ws;
    float* h2    = h1 + BND;
    float* qb    = h2 + BND;
    float* kb    = qb + BND;
    float* vb2   = kb + BND;
    float* ob    = vb2 + BND;
    float* att   = ob + BND;
    float* hmid  = att + BND;
    float* maskb = hmid + BNF;

    const int M = Bb * Nn;                     // 8192 rows

    // 1) fused LN(x) with both parameter sets
    ln2_kernel<<<M, 256, 0, stream>>>(x, g1, be1, g2, be2, h1, h2);

    // 2) Q/K/V projections: (8192/64)*(768/32) = 3072 waves / 8 = 384 blocks
    gemm_kernel<0><<<384, 256, 0, stream>>>(h1, Wq, nullptr, nullptr, nullptr, qb,  M, Dd, Dd);
    gemm_kernel<0><<<384, 256, 0, stream>>>(h1, Wk, nullptr, nullptr, nullptr, kb,  M, Dd, Dd);
    gemm_kernel<0><<<384, 256, 0, stream>>>(h1, Wv, bv,      nullptr, nullptr, vb2, M, Dd, Dd);

    // 3) fused attention: B*H*(N/16) = 6144 single-wave blocks
    attn_kernel<<<Bb * Hh * (Nn / 16), 32, 0, stream>>>(qb, kb, vb2, sp, ed, ob);

    // 4) output projection + bo + residual x
    gemm_kernel<1><<<384, 256, 0, stream>>>(ob, Wo, bo, x, nullptr, att, M, Dd, Dd);

    // 5) pad mask from att
    mask_kernel<<<M, 256, 0, stream>>>(att, maskb);

    // 6) FFN: gelu(h2 @ W1 + bf1): (128*96)/8 = 1536 blocks
    gemm_kernel<2><<<1536, 256, 0, stream>>>(h2, W1, bf1, nullptr, nullptr, hmid, M, Ff, Dd);

    // 7) hmid @ W2 + bf2 + att, zero masked rows -> final output
    gemm_kernel<3><<<384, 256, 0, stream>>>(hmid, W2, bf2, att, maskb, out, M, Dd, Ff);
}